// BlockTopK_78357383348740
// MI455X (gfx1250) — compile-verified
//
#include <hip/hip_runtime.h>

// blocktopk(x, k=2, block_size=4) mask kernel for MI455X (gfx1250).
//
// Memory-bound streaming transform: 256 MiB in + 256 MiB out @ 23.3 TB/s
// => ~23 us floor. Compute is 6 fp compares + a few carry-adds per block of 4
// (~100x below the VALU roof), so the design targets bandwidth only:
//   - one float4 (b128) per lane per access: 512 B contiguous per wave32
//   - non-temporal TH hints both directions (touch-once, 512 MiB >> 192 MB L2)
//   - unconditional fast path issues all UNROLL b128 loads back-to-back so the
//     wave keeps 4 loads (2 KB) in flight instead of load->wait->compute chains
//   - no LDS, no WMMA (this op contains no matrix math).

typedef __attribute__((ext_vector_type(4))) float v4f;

#define TPB    256   // 8 wave32s per workgroup
#define UNROLL 4     // 4 independent b128 loads in flight per thread

__device__ __forceinline__ v4f top2_mask(v4f v)
{
    const float a = v.x, b = v.y, c = v.z, d = v.w;

    // 6 pairwise compares; complements give stable top-k ranks:
    // rank(i) = #{j<i: x[j] >= x[i]} + #{j>i: x[j] > x[i]}
    const bool ab = (b > a), ac = (c > a), ad = (d > a);
    const bool bc = (c > b), bd = (d > b), cd = (d > c);

    const int ra =  (int)ab +  (int)ac +  (int)ad;
    const int rb = (int)!ab +  (int)bc +  (int)bd;
    const int rc = (int)!ac + (int)!bc +  (int)cd;
    const int rd = (int)!ad + (int)!bd + (int)!cd;

    v4f m;
    m.x = (ra < 2) ? 1.0f : 0.0f;
    m.y = (rb < 2) ? 1.0f : 0.0f;
    m.z = (rc < 2) ? 1.0f : 0.0f;
    m.w = (rd < 2) ? 1.0f : 0.0f;
    return m;
}

__global__ __launch_bounds__(TPB) void blocktopk_mask_kernel(
    const v4f* __restrict__ in, v4f* __restrict__ out, int nvec, int span)
{
    const int tid = blockIdx.x * TPB + threadIdx.x;

    if (tid + (UNROLL - 1) * span < nvec) {
        // Fast path (taken by every thread when the grid divides nvec, as it
        // does for 8192x8192): no per-item branches -> all UNROLL
        // global_load_b128 issued back-to-back, waits staggered by use.
        v4f v[UNROLL];
#pragma unroll
        for (int u = 0; u < UNROLL; ++u)
            v[u] = __builtin_nontemporal_load(in + tid + u * span);

#pragma unroll
        for (int u = 0; u < UNROLL; ++u)
            __builtin_nontemporal_store(top2_mask(v[u]), out + tid + u * span);
    } else {
        // Guarded tail path for sizes the grid doesn't divide evenly.
#pragma unroll
        for (int u = 0; u < UNROLL; ++u) {
            const int i = tid + u * span;
            if (i < nvec) {
                v4f v = __builtin_nontemporal_load(in + i);
                __builtin_nontemporal_store(top2_mask(v), out + i);
            }
        }
    }
}

extern "C" void kernel_launch(void* const* d_in, const int* in_sizes, int n_in,
                              void* d_out, int out_size, void* d_ws, size_t ws_size,
                              hipStream_t stream)
{
    (void)n_in; (void)out_size; (void)d_ws; (void)ws_size;

    const v4f* in  = (const v4f*)d_in[0];
    v4f*       out = (v4f*)d_out;

    const int n    = in_sizes[0];      // 8192*8192 = 67,108,864
    const int nvec = n / 4;            // 16,777,216 float4 blocks

    const int blocks = (nvec + TPB * UNROLL - 1) / (TPB * UNROLL);  // 16384
    const int span   = blocks * TPB;   // grid span between a thread's UNROLL items

    blocktopk_mask_kernel<<<dim3(blocks), dim3(TPB), 0, stream>>>(in, out, nvec, span);
}